// RegionProposalNetwork_30794915512675
// MI455X (gfx1250) — compile-verified
//
#include <hip/hip_runtime.h>
#include <stdint.h>

// ---------------- problem constants ----------------
#define NIMG 8
#define AA   15
#define HH   200
#define WW   320
#define HW   (HH*WW)          // 64000
#define TOT  (AA*HW)          // 960000 anchors per image
#define PRE  2000
#define POST 1000
#define CANDN 4096
#define SLOTS 2048
#define NMS_T 0.7f
#define BCLIP 4.135166556742356f   // log(1000/16)

// ---------------- workspace layout (bytes) ----------------
#define OFF_HIST   0u
#define SZ_HIST    (NIMG*65536u*4u)                 // 2,097,152
#define OFF_CNT    (OFF_HIST + SZ_HIST)
#define OFF_THR    (OFF_CNT + 32u)
#define OFF_CAND   (OFF_CNT + 64u)                  // 8-byte aligned
#define SZ_CAND    (NIMG*CANDN*8u)                  // 262,144
#define OFF_BOXT   (OFF_CAND + SZ_CAND)             // float4 tmp boxes
#define SZ_BOXT    (NIMG*SLOTS*16u)
#define OFF_BOXS   (OFF_BOXT + SZ_BOXT)             // sorted boxes
#define OFF_SCS    (OFF_BOXS + SZ_BOXT)             // sorted masked scores
#define SZ_SCS     (NIMG*SLOTS*4u)
#define OFF_MASK   (OFF_SCS + SZ_SCS)               // u16[img][2048][128]
#define SZ_MASK    (NIMG*SLOTS*128u*2u)
#define CLEAR_BYTES (OFF_CAND + SZ_CAND)            // hist+cnt+thr+cand

typedef __attribute__((ext_vector_type(2))) float v2f;
typedef __attribute__((ext_vector_type(8))) float v8f;
typedef __attribute__((ext_vector_type(4))) int   v4i;
typedef __attribute__((address_space(1))) v4i* as1_v4i;   // global int4*
typedef __attribute__((address_space(3))) v4i* as3_v4i;   // LDS int4*

#if defined(__has_builtin)
#if __has_builtin(__builtin_amdgcn_global_load_async_to_lds_b128)
#define HAVE_ASYNC_LDS 1
#endif
#if __has_builtin(__builtin_amdgcn_wmma_f32_16x16x4_f32)
#define HAVE_WMMA_F32 1
#endif
#endif

__device__ __forceinline__ uint32_t ballot32(bool p) {
#if defined(__has_builtin) && __has_builtin(__builtin_amdgcn_ballot_w32)
  return __builtin_amdgcn_ballot_w32(p);
#else
  return (uint32_t)__ballot(p);
#endif
}

__device__ __forceinline__ uint32_t sortkey(float f) {
  uint32_t u = __float_as_uint(f);
  return (u & 0x80000000u) ? ~u : (u | 0x80000000u);
}
__device__ __forceinline__ float unsortkey(uint32_t k) {
  return (k & 0x80000000u) ? __uint_as_float(k ^ 0x80000000u) : __uint_as_float(~k);
}

// ===== K1: 16-bit-bin histogram of sortable objectness keys (per image) =====
__global__ void rpn_hist(const float* __restrict__ cls, uint32_t* __restrict__ hist) {
  int gid = blockIdx.x * blockDim.x + threadIdx.x;
  if (gid >= NIMG * TOT) return;
  int img = gid / TOT;
  uint32_t key = sortkey(cls[gid]);
  atomicAdd(&hist[img * 65536 + (key >> 16)], 1u);
}

// ===== K2: find threshold bin B (cum count of bins >= B first reaches PRE) =====
__global__ void rpn_thresh(const uint32_t* __restrict__ hist, uint32_t* __restrict__ thr) {
  int img = blockIdx.x;
  const uint32_t* h = hist + img * 65536;
  __shared__ uint32_t sums[256];
  __shared__ uint32_t before[256];
  __shared__ int foundB;
  int t = threadIdx.x;
  if (t == 0) foundB = -1;
  int hi = 65535 - t * 256;                 // descending chunks of 256 bins
  uint32_t s = 0;
  for (int k = 0; k < 256; ++k) s += h[hi - k];
  sums[t] = s;
  __syncthreads();
  if (t == 0) { uint32_t c = 0; for (int i = 0; i < 256; ++i) { before[i] = c; c += sums[i]; } }
  __syncthreads();
  uint32_t cum = before[t];
  if (cum < PRE && cum + sums[t] >= PRE) {  // exactly one thread's chunk crosses
    for (int k = 0; k < 256; ++k) {
      uint32_t c = h[hi - k];
      if (cum < PRE && cum + c >= PRE) { foundB = hi - k; break; }
      cum += c;
    }
  }
  __syncthreads();
  if (t == 0) thr[img] = (foundB < 0) ? 0u : (uint32_t)foundB;
}

// ===== K3: compact all elements with bin >= B into candidate buffer =====
__global__ void rpn_compact(const float* __restrict__ cls, const uint32_t* __restrict__ thr,
                            uint32_t* __restrict__ cnt, unsigned long long* __restrict__ cand) {
  int gid = blockIdx.x * blockDim.x + threadIdx.x;
  if (gid >= NIMG * TOT) return;
  int img = gid / TOT;
  int off = gid - img * TOT;                // layout (a, h, w)
  uint32_t key = sortkey(cls[gid]);
  if ((key >> 16) >= thr[img]) {
    uint32_t pos = atomicAdd(&cnt[img], 1u);
    if (pos < CANDN) {
      int a = off / HW;
      int hw = off - a * HW;
      uint32_t idx = (uint32_t)(hw * AA + a); // reference flat anchor index
      cand[(size_t)img * CANDN + pos] = ((unsigned long long)key << 32) | (uint32_t)(~idx);
    }
  }
}

// ---- bitonic sort, descending, N power of two, NT threads ----
template <int N, int NT>
__device__ __forceinline__ void bitonic_desc(unsigned long long* key, int t) {
  for (int ksz = 2; ksz <= N; ksz <<= 1) {
    for (int j = ksz >> 1; j > 0; j >>= 1) {
      __syncthreads();
      for (int i = t; i < N; i += NT) {
        int ixj = i ^ j;
        if (ixj > i) {
          unsigned long long a = key[i], b = key[ixj];
          bool desc = ((i & ksz) == 0);
          if (desc ? (a < b) : (a > b)) { key[i] = b; key[ixj] = a; }
        }
      }
    }
  }
  __syncthreads();
}

// ===== K4: sort candidates, decode top-2000, mask, re-sort by masked score =====
__global__ __launch_bounds__(1024) void rpn_sortdecode(
    const float* __restrict__ cls, const float* __restrict__ reg,
    const float* __restrict__ anch, const int* __restrict__ ih, const int* __restrict__ iw,
    const unsigned long long* __restrict__ cand,
    float4* __restrict__ boxT, float4* __restrict__ boxS, float* __restrict__ scoreS) {
  int img = blockIdx.x;
  __shared__ unsigned long long key[CANDN];
  __shared__ float scm[SLOTS];
  int t = threadIdx.x;
  for (int k = t; k < CANDN; k += 1024) key[k] = cand[(size_t)img * CANDN + k];
  bitonic_desc<CANDN, 1024>(key, t);        // exact top-k with index tiebreak

  float fw = (float)iw[0], fh = (float)ih[0];
  for (int s = t; s < SLOTS; s += 1024) {
    unsigned long long p = key[s];
    uint32_t k32 = (uint32_t)(p >> 32);
    float4 box = {0.f, 0.f, 0.f, 0.f};
    float sm = -1.0f;
    if (k32 != 0u && s < PRE) {             // k32==0 -> padding slot
      uint32_t idx = ~(uint32_t)p;
      int a = (int)(idx % AA);
      int hw = (int)(idx / AA);
      float raw = cls[(size_t)img * TOT + (size_t)a * HW + hw];
      float score = 1.0f / (1.0f + __expf(-raw));
      const float* an = anch + (size_t)idx * 4;
      float ax0 = an[0], ay0 = an[1], ax1 = an[2], ay1 = an[3];
      float wa = ax1 - ax0, ha = ay1 - ay0;
      float cxa = ax0 + 0.5f * wa, cya = ay0 + 0.5f * ha;
      size_t rb = ((size_t)img * 4 * AA + (size_t)a * 4) * HW + hw;
      float dx = reg[rb], dy = reg[rb + HW];
      float dw = fminf(reg[rb + 2 * (size_t)HW], BCLIP);
      float dh = fminf(reg[rb + 3 * (size_t)HW], BCLIP);
      float cx = dx * wa + cxa, cy = dy * ha + cya;
      float w = wa * __expf(dw), h = ha * __expf(dh);
      float x0 = fminf(fmaxf(cx - 0.5f * w, 0.f), fw);
      float y0 = fminf(fmaxf(cy - 0.5f * h, 0.f), fh);
      float x1 = fminf(fmaxf(cx + 0.5f * w, 0.f), fw);
      float y1 = fminf(fmaxf(cy + 0.5f * h, 0.f), fh);
      box = make_float4(x0, y0, x1, y1);
      bool valid = ((x1 - x0) >= 1.0f) && ((y1 - y0) >= 1.0f) && (score >= 0.0f);
      sm = valid ? score : -1.0f;
    }
    boxT[(size_t)img * SLOTS + s] = box;
    scm[s] = sm;
  }
  __syncthreads();
  for (int s = t; s < SLOTS; s += 1024)     // stable-desc sort by masked score
    key[s] = ((unsigned long long)sortkey(scm[s]) << 32) | (uint32_t)(SLOTS - 1 - s);
  bitonic_desc<SLOTS, 1024>(key, t);
  for (int s = t; s < SLOTS; s += 1024) {
    unsigned long long p = key[s];
    int sorig = (SLOTS - 1) - (int)(uint32_t)(p & 0xFFFFFFFFu);
    boxS[(size_t)img * SLOTS + s] = boxT[(size_t)img * SLOTS + sorig];
    scoreS[(size_t)img * SLOTS + s] = unsortkey((uint32_t)(p >> 32));
  }
}

// ===== K5: suppression bitmask. WMMA computes area_i+area_j outer-sum tiles;
//           boxes staged to LDS via async global->LDS copy. =====
__global__ __launch_bounds__(256) void rpn_mask(const float4* __restrict__ boxS,
                                                uint16_t* __restrict__ mask) {
  int img = blockIdx.y;
  int iT  = blockIdx.x;                     // 128 i-tiles of 16 rows
  __shared__ float4 B[SLOTS];               // 32 KB box table
  const float4* src = boxS + (size_t)img * SLOTS;
  for (int k = 0; k < 8; ++k) {
    int e = threadIdx.x + k * 256;
#ifdef HAVE_ASYNC_LDS
    __builtin_amdgcn_global_load_async_to_lds_b128(
        (as1_v4i)(src + e), (as3_v4i)(&B[e]), 0, 0);
#else
    B[e] = src[e];
#endif
  }
#ifdef HAVE_ASYNC_LDS
#if defined(__has_builtin) && __has_builtin(__builtin_amdgcn_s_wait_asynccnt)
  __builtin_amdgcn_s_wait_asynccnt(0);
#else
  asm volatile("s_wait_asynccnt 0" ::: "memory");
#endif
#endif
  __syncthreads();

  int lane = threadIdx.x & 31;
  int wv   = threadIdx.x >> 5;
  int i0   = iT * 16;
  float areaI = 0.f;
  if (lane < 16) { float4 b = B[i0 + lane]; areaI = (b.z - b.x) * (b.w - b.y); }
  v2f aop; aop[0] = (lane < 16) ? areaI : 0.f; aop[1] = (lane < 16) ? 1.0f : 0.f;
  uint16_t* mrow = mask + (size_t)img * SLOTS * 128;

  for (int jT = wv; jT < 128; jT += 8) {
    int j0 = jT * 16;
    float areaJ = 0.f;
    if (lane < 16) { float4 b = B[j0 + lane]; areaJ = (b.z - b.x) * (b.w - b.y); }
#ifdef HAVE_WMMA_F32
    v2f bop; bop[0] = (lane < 16) ? 1.0f : 0.f; bop[1] = (lane < 16) ? areaJ : 0.f;
    v8f c = {0.f, 0.f, 0.f, 0.f, 0.f, 0.f, 0.f, 0.f};
    // D[m][n] = area_i[m]*1 + 1*area_j[n]   (16x16x4 f32 WMMA, K=2..3 zeroed)
    v8f S = __builtin_amdgcn_wmma_f32_16x16x4_f32(false, aop, false, bop,
                                                  (short)0, c, false, false);
#endif
    int n = lane & 15;
    int j = j0 + n;
    float4 bj = B[j];
    float areaJn = (bj.z - bj.x) * (bj.w - bj.y);
    (void)areaJn;
    int mb = (lane >> 4) << 3;              // lanes 16-31 hold rows m+8
    for (int v = 0; v < 8; ++v) {
      int i = i0 + mb + v;
      float4 bi = B[i];
      float ix = fmaxf(fminf(bi.z, bj.z) - fmaxf(bi.x, bj.x), 0.f);
      float iy = fmaxf(fminf(bi.w, bj.w) - fmaxf(bi.y, bj.y), 0.f);
      float inter = ix * iy;
#ifdef HAVE_WMMA_F32
      float Sv = S[v];
#else
      float Sv = (bi.z - bi.x) * (bi.w - bi.y) + areaJn;
#endif
      // iou > T  <=>  inter*(1+T) > T*(area_i+area_j)   (exact rearrangement)
      bool pred = (inter * (1.0f + NMS_T) > NMS_T * Sv) && (j > i);
      uint32_t b32 = ballot32(pred);
      if (lane == 0) {
        mrow[(size_t)(i0 + v) * 128 + jT]     = (uint16_t)(b32 & 0xFFFFu);
        mrow[(size_t)(i0 + v + 8) * 128 + jT] = (uint16_t)(b32 >> 16);
      }
    }
  }
}

// ===== K6: serial greedy scan (one wave per image) + compaction to output =====
__global__ __launch_bounds__(256) void rpn_scanout(
    const float4* __restrict__ boxS, const float* __restrict__ scoreS,
    const uint16_t* __restrict__ mask, float* __restrict__ outB, float* __restrict__ outS) {
  int img = blockIdx.x;
  __shared__ float sc[SLOTS];
  __shared__ volatile uint32_t remv[64];
  __shared__ volatile uint8_t kf[SLOTS];
  __shared__ uint32_t psum[256];
  int t = threadIdx.x;
  for (int s = t; s < SLOTS; s += 256) { sc[s] = scoreS[(size_t)img * SLOTS + s]; kf[s] = 0; }
  if (t < 64) remv[t] = 0;
  __syncthreads();
  if (t < 32) {
    const uint32_t* mrow0 = (const uint32_t*)(mask + (size_t)img * SLOTS * 128);
    for (int i = 0; i < PRE; ++i) {
      uint32_t r = remv[i >> 5];
      bool cond = (sc[i] >= 0.0f) && (((r >> (i & 31)) & 1u) == 0u);
      if (cond) {
        if (t == 0) kf[i] = 1;
        const uint32_t* row = mrow0 + (size_t)i * 64;
        remv[t]      |= row[t];
        remv[t + 32] |= row[t + 32];
      }
    }
  }
  __syncthreads();
  uint32_t local = 0;
  int base = t * 8;
  for (int k = 0; k < 8; ++k) local += kf[base + k];
  psum[t] = local;
  __syncthreads();
  if (t == 0) { uint32_t c = 0; for (int i = 0; i < 256; ++i) { uint32_t v = psum[i]; psum[i] = c; c += v; } }
  __syncthreads();
  uint32_t pos = psum[t];
  for (int k = 0; k < 8; ++k) {
    int s = base + k;
    if (kf[s]) {
      if (pos < POST) {
        float4 b = boxS[(size_t)img * SLOTS + s];
        float* ob = outB + ((size_t)img * POST + pos) * 4;
        ob[0] = b.x; ob[1] = b.y; ob[2] = b.z; ob[3] = b.w;
        outS[(size_t)img * POST + pos] = sc[s];
      }
      pos++;
    }
  }
}

extern "C" void kernel_launch(void* const* d_in, const int* in_sizes, int n_in,
                              void* d_out, int out_size, void* d_ws, size_t ws_size,
                              hipStream_t stream) {
  const float* cls  = (const float*)d_in[0];
  const float* reg  = (const float*)d_in[1];
  const float* anch = (const float*)d_in[2];
  const int*   ih   = (const int*)d_in[3];
  const int*   iw   = (const int*)d_in[4];

  char* ws = (char*)d_ws;
  uint32_t*           hist = (uint32_t*)(ws + OFF_HIST);
  uint32_t*           cnt  = (uint32_t*)(ws + OFF_CNT);
  uint32_t*           thr  = (uint32_t*)(ws + OFF_THR);
  unsigned long long* cand = (unsigned long long*)(ws + OFF_CAND);
  float4*             boxT = (float4*)(ws + OFF_BOXT);
  float4*             boxS = (float4*)(ws + OFF_BOXS);
  float*              scS  = (float*)(ws + OFF_SCS);
  uint16_t*           msk  = (uint16_t*)(ws + OFF_MASK);

  (void)hipMemsetAsync(ws, 0, (size_t)CLEAR_BYTES, stream);
  (void)hipMemsetAsync(d_out, 0, (size_t)out_size * sizeof(float), stream);

  int nElem = NIMG * TOT;
  rpn_hist<<<(nElem + 255) / 256, 256, 0, stream>>>(cls, hist);
  rpn_thresh<<<NIMG, 256, 0, stream>>>(hist, thr);
  rpn_compact<<<(nElem + 255) / 256, 256, 0, stream>>>(cls, thr, cnt, cand);
  rpn_sortdecode<<<NIMG, 1024, 0, stream>>>(cls, reg, anch, ih, iw, cand, boxT, boxS, scS);
  rpn_mask<<<dim3(128, NIMG), 256, 0, stream>>>(boxS, msk);
  float* outB = (float*)d_out;
  float* outS = outB + (size_t)NIMG * POST * 4;
  rpn_scanout<<<NIMG, 256, 0, stream>>>(boxS, scS, msk, outB, outS);
}